// S4DLayer_33028298506610
// MI455X (gfx1250) — compile-verified
//
#include <hip/hip_runtime.h>
#include <hip/hip_bf16.h>

// S4D layer on gfx1250: chunked state-space algorithm, 3x 64x64x64 f32 WMMA
// GEMMs per (b,h) block + tiny 64-step complex scan. u transposed to (B,H,L)
// in d_ws, processed in place, transposed back.

#define Hh 512
#define N2 32
#define Bb 8
#define Ll 4096
#define Tc 64        // chunk length
#define NC 64        // number of chunks (Ll / Tc)
#define LDW 65       // padded LDS row stride (floats)

typedef __attribute__((ext_vector_type(2))) float v2f;
typedef __attribute__((ext_vector_type(8))) float v8f;

// ---------------- transpose (B, M, N) -> (B, N, M) ----------------
__global__ void __launch_bounds__(256)
s4d_transpose(const float* __restrict__ in, float* __restrict__ out, int M, int N) {
  __shared__ float tile[32][33];
  const int b  = blockIdx.z;
  const int n0 = blockIdx.x * 32;
  const int m0 = blockIdx.y * 32;
  const int tx = threadIdx.x, ty = threadIdx.y;   // 32 x 8
  const float* inb  = in  + (size_t)b * M * N;
  float*       outb = out + (size_t)b * M * N;
#pragma unroll
  for (int i = 0; i < 32; i += 8)
    tile[ty + i][tx] = inb[(size_t)(m0 + ty + i) * N + (n0 + tx)];
  __syncthreads();
#pragma unroll
  for (int i = 0; i < 32; i += 8)
    outb[(size_t)(n0 + ty + i) * M + (m0 + tx)] = tile[tx][ty + i];
}

// ---------------- main fused S4D block kernel ----------------
// one block = one (b,h); ut is (B,H,L), updated in place with y.
__global__ void __launch_bounds__(256)
s4d_main(float* __restrict__ ut, const float* __restrict__ C,
         const float* __restrict__ D, const float* __restrict__ log_delta,
         const float* __restrict__ log_A_real, const float* __restrict__ A_imag) {
  extern __shared__ float sm[];
  float* u_loc = sm;                 // 64 x LDW : u chunks Uc[c][j]
  float* T2    = sm + 4160;          // 64 x LDW : T2[j][i] = K[i-j]
  float* VmT   = sm + 8320;          // 64 x LDW : VmT[j][2n(+1)] = Re/Im(w^{63-j})
  float* MmT   = sm + 12480;         // 64 x LDW : MmT[2n(+1)][i] = +-2Re/Im(BC w^{i+1})
  float* varr  = sm + 16640;         // 64 x LDW : v(c) then S(c)
  float* Yi    = sm + 20800;         // 64 x LDW : y chunks
  float* s_ar  = sm + 24960;         // 32: Re(dtA)
  float* s_ai  = s_ar + 32;          // 32: Im(dtA)
  float* s_br  = s_ar + 64;          // 32: Re(BC)
  float* s_bi  = s_ar + 96;          // 32: Im(BC)
  float* s_zr  = s_ar + 128;         // 32: Re(w^64)
  float* s_zi  = s_ar + 160;         // 32: Im(w^64)
  float* s_K   = s_ar + 192;         // 64: K[m]

  const int tid = threadIdx.x;
  const int h   = blockIdx.x & (Hh - 1);
  const int b   = blockIdx.x >> 9;
  float* ubase  = ut + ((size_t)b * Hh + h) * Ll;

  // stage u[b,h,:] into LDS (coalesced), rows padded to LDW
  for (int idx = tid; idx < Ll; idx += 256)
    u_loc[(idx >> 6) * LDW + (idx & 63)] = ubase[idx];

  // per-mode constants
  if (tid < N2) {
    const int n = tid;
    float dt   = expf(log_delta[h]);
    float Are  = -expf(log_A_real[h * N2 + n]);
    float Aim  = A_imag[h * N2 + n];
    float dar  = Are * dt, dai = Aim * dt;           // dtA
    float ew   = expf(dar);
    float swv, cw; sincosf(dai, &swv, &cw);
    float wr = ew * cw, wi = ew * swv;               // w = exp(dtA)
    float d2 = Are * Are + Aim * Aim;
    float xr = wr - 1.0f, xi = wi;                   // (w-1)/A
    float qr = (xr * Are + xi * Aim) / d2;
    float qi = (xi * Are - xr * Aim) / d2;
    float Cr = C[(h * N2 + n) * 2 + 0];
    float Ci = C[(h * N2 + n) * 2 + 1];
    float ez = expf(64.0f * dar);                    // z = w^64
    float sz, cz; sincosf(64.0f * dai, &sz, &cz);
    s_ar[n] = dar;  s_ai[n] = dai;
    s_br[n] = Cr * qr - Ci * qi;
    s_bi[n] = Cr * qi + Ci * qr;
    s_zr[n] = ez * cz;  s_zi[n] = ez * sz;
  }
  __syncthreads();

  // K[m] = 2 sum_n Re(BC_n w_n^m), m = 0..63
  if (tid < Tc) {
    const float m = (float)tid;
    float acc = 0.0f;
    for (int n = 0; n < N2; ++n) {
      float em = expf(m * s_ar[n]);
      float smv, cm; sincosf(m * s_ai[n], &smv, &cm);
      acc += s_br[n] * (em * cm) - s_bi[n] * (em * smv);
    }
    s_K[tid] = 2.0f * acc;
  }
  __syncthreads();

  // build B-matrices
  for (int idx = tid; idx < 4096; idx += 256) {      // T2[j][i] = K[i-j] (i>=j)
    int j = idx >> 6, i = idx & 63;
    T2[j * LDW + i] = (i >= j) ? s_K[i - j] : 0.0f;
  }
  for (int idx = tid; idx < 2048; idx += 256) {      // VmT[j][2n(+1)] = w^{63-j}
    int j = idx >> 5, n = idx & 31;
    float m = (float)(63 - j);
    float em = expf(m * s_ar[n]);
    float smv, cm; sincosf(m * s_ai[n], &smv, &cm);
    VmT[j * LDW + 2 * n]     = em * cm;
    VmT[j * LDW + 2 * n + 1] = em * smv;
  }
  for (int idx = tid; idx < 2048; idx += 256) {      // MmT rows 2n,2n+1
    int n = idx >> 6, i = idx & 63;
    float m = (float)(i + 1);
    float em = expf(m * s_ar[n]);
    float smv, cm; sincosf(m * s_ai[n], &smv, &cm);
    float wrp = em * cm, wip = em * smv;
    float ar2 = s_br[n] * wrp - s_bi[n] * wip;       // BC * w^{i+1}
    float ai2 = s_br[n] * wip + s_bi[n] * wrp;
    MmT[(2 * n) * LDW + i]     =  2.0f * ar2;
    MmT[(2 * n + 1) * LDW + i] = -2.0f * ai2;
  }
  __syncthreads();

  const int wave = tid >> 5, lane = tid & 31;
  const int half = lane >> 4, lo = lane & 15;

  // GEMM1: Yi = Uc @ T2   |   GEMM2: varr = Uc @ VmT   (32 tiles over 8 waves)
  for (int t = wave; t < 32; t += 8) {
    const float* Bm = (t < 16) ? T2 : VmT;
    float*       Om = (t < 16) ? Yi : varr;
    const int tt = t & 15;
    const int tm = tt >> 2, tn = tt & 3;
    v8f c = {};
    for (int k0 = 0; k0 < 64; k0 += 4) {
      v2f a, bb;
      a.x  = u_loc[(tm * 16 + lo) * LDW + k0 + 2 * half];
      a.y  = u_loc[(tm * 16 + lo) * LDW + k0 + 2 * half + 1];
      bb.x = Bm[(k0 + 2 * half) * LDW + tn * 16 + lo];
      bb.y = Bm[(k0 + 2 * half + 1) * LDW + tn * 16 + lo];
      c = __builtin_amdgcn_wmma_f32_16x16x4_f32(false, a, false, bb,
                                                (short)0, c, false, false);
    }
#pragma unroll
    for (int r = 0; r < 8; ++r)
      Om[(tm * 16 + r + 8 * half) * LDW + tn * 16 + lo] = c[r];
  }
  __syncthreads();

  // chunk-level complex scan: varr[c] <- S(c);  S(c+1) = z*S(c) + v(c)
  if (tid < N2) {
    const int n = tid;
    const float zr = s_zr[n], zi = s_zi[n];
    float sr = 0.0f, si = 0.0f;
    for (int c = 0; c < NC; ++c) {
      float vr = varr[c * LDW + 2 * n];
      float vi = varr[c * LDW + 2 * n + 1];
      varr[c * LDW + 2 * n]     = sr;
      varr[c * LDW + 2 * n + 1] = si;
      float nsr = zr * sr - zi * si + vr;
      float nsi = zr * si + zi * sr + vi;
      sr = nsr; si = nsi;
    }
  }
  __syncthreads();

  // GEMM3: Yi += S @ MmT  (accumulate via C operand)
  for (int t = wave; t < 16; t += 8) {
    const int tm = t >> 2, tn = t & 3;
    v8f c;
#pragma unroll
    for (int r = 0; r < 8; ++r)
      c[r] = Yi[(tm * 16 + r + 8 * half) * LDW + tn * 16 + lo];
    for (int k0 = 0; k0 < 64; k0 += 4) {
      v2f a, bb;
      a.x  = varr[(tm * 16 + lo) * LDW + k0 + 2 * half];
      a.y  = varr[(tm * 16 + lo) * LDW + k0 + 2 * half + 1];
      bb.x = MmT[(k0 + 2 * half) * LDW + tn * 16 + lo];
      bb.y = MmT[(k0 + 2 * half + 1) * LDW + tn * 16 + lo];
      c = __builtin_amdgcn_wmma_f32_16x16x4_f32(false, a, false, bb,
                                                (short)0, c, false, false);
    }
#pragma unroll
    for (int r = 0; r < 8; ++r)
      Yi[(tm * 16 + r + 8 * half) * LDW + tn * 16 + lo] = c[r];
  }
  __syncthreads();

  // y = Yi + D*u, written back in place into ut (coalesced)
  const float Dh = D[h];
  for (int idx = tid; idx < Ll; idx += 256) {
    int cI = idx >> 6, j = idx & 63;
    ubase[idx] = Yi[cI * LDW + j] + Dh * u_loc[cI * LDW + j];
  }
}

extern "C" void kernel_launch(void* const* d_in, const int* in_sizes, int n_in,
                              void* d_out, int out_size, void* d_ws, size_t ws_size,
                              hipStream_t stream) {
  (void)in_sizes; (void)n_in; (void)out_size; (void)ws_size;
  const float* u   = (const float*)d_in[0];
  const float* C   = (const float*)d_in[1];
  const float* D   = (const float*)d_in[2];
  const float* ld  = (const float*)d_in[3];
  const float* lar = (const float*)d_in[4];
  const float* aim = (const float*)d_in[5];
  float* out = (float*)d_out;
  float* ut  = (float*)d_ws;                       // B*H*L floats = 64 MB

  const size_t smem = (size_t)(25216) * sizeof(float);  // ~98.5 KB
  dim3 tb(32, 8, 1);

  // u (B,L,H) -> ut (B,H,L)
  s4d_transpose<<<dim3(Hh / 32, Ll / 32, Bb), tb, 0, stream>>>(u, ut, Ll, Hh);
  // fused S4D, y written in place into ut
  s4d_main<<<dim3(Bb * Hh, 1, 1), dim3(256, 1, 1), smem, stream>>>(ut, C, D, ld, lar, aim);
  // ut (B,H,L) -> out (B,L,H)
  s4d_transpose<<<dim3(Ll / 32, Hh / 32, Bb), tb, 0, stream>>>(ut, out, Hh, Ll);
}